// MinibatchDiscriminatin_82600811037305
// MI455X (gfx1250) — compile-verified
//
#include <hip/hip_runtime.h>

// out[n,f] = sum_m exp( - sum_d |x[n,f,d] - x[m,f,d]| ),  x: (256, 128, 32) f32
// VALU-bound (L1 distance is not bilinear -> no WMMA formulation exists).
// CDNA5 path used: async global->LDS staging (ASYNCcnt) + wave32 VALU + v_exp_f32.

#define N_ 256
#define F_ 128
#define D_ 32
#define NTILE 128           // rows of n per block
#define NF4 (D_ / 4)        // 8 float4 per row
#define SLICE_F4 (N_ * NF4) // 2048 float4 = 32 KB LDS per block

#if defined(__gfx1250__)
typedef int v4i_t __attribute__((ext_vector_type(4)));
typedef __attribute__((address_space(1))) v4i_t* gptr_v4i;
typedef __attribute__((address_space(3))) v4i_t* lptr_v4i;
#endif

__global__ __launch_bounds__(NTILE) void MinibatchDisc_kernel(
    const float* __restrict__ x, float* __restrict__ out) {
  const int f   = blockIdx.x;               // feature 0..127
  const int tid = threadIdx.x;              // 0..127
  const int n   = blockIdx.y * NTILE + tid; // 0..255

  __shared__ float4 s[SLICE_F4]; // x[:, f, :] as 256 rows x 8 float4

  // ---- Stage the whole feature slice into LDS -----------------------------
  // float4 index i in [0,2048): row m = i>>3, chunk j = i&7 (8 float4 / row)
  // global float offset of chunk: (m*F_ + f)*D_ + 4*j
#if defined(__gfx1250__) && __has_builtin(__builtin_amdgcn_global_load_async_to_lds_b128)
  {
    const char* gbase = (const char*)x;
#pragma unroll
    for (int k = 0; k < SLICE_F4 / NTILE; ++k) { // 16 iterations
      int i = tid + k * NTILE;
      int m = i >> 3;
      int j = i & 7;
      size_t goff = ((size_t)m * (F_ * D_) + (size_t)f * D_ + 4 * j) * sizeof(float);
      __builtin_amdgcn_global_load_async_to_lds_b128(
          (gptr_v4i)(void*)(gbase + goff),
          (lptr_v4i)&s[i],
          /*imm offset*/ 0, /*cpol*/ 0);
    }
#if __has_builtin(__builtin_amdgcn_s_wait_asynccnt)
    __builtin_amdgcn_s_wait_asynccnt(0);
#else
    asm volatile("s_wait_asynccnt 0" ::: "memory");
#endif
  }
#else
  {
    const float4* xg = (const float4*)x;
#pragma unroll
    for (int k = 0; k < SLICE_F4 / NTILE; ++k) {
      int i = tid + k * NTILE;
      int m = i >> 3;
      int j = i & 7;
      s[i] = xg[(size_t)m * (F_ * D_ / 4) + (size_t)f * (D_ / 4) + j];
    }
  }
#endif
  __syncthreads();

  // ---- Cache own row n in registers (32 VGPRs) ----------------------------
  float4 xn[NF4];
#pragma unroll
  for (int j = 0; j < NF4; ++j) xn[j] = s[n * NF4 + j];

  // ---- Sweep all m rows; LDS reads are lane-uniform -> broadcast ----------
  float acc = 0.0f;
  for (int m = 0; m < N_; ++m) {
    const float4* row = &s[m * NF4];
    float d0 = 0.0f, d1 = 0.0f, d2 = 0.0f, d3 = 0.0f; // 4 chains for ILP
#pragma unroll
    for (int j = 0; j < NF4; ++j) {
      float4 v = row[j];
      d0 += __builtin_fabsf(xn[j].x - v.x);
      d1 += __builtin_fabsf(xn[j].y - v.y);
      d2 += __builtin_fabsf(xn[j].z - v.z);
      d3 += __builtin_fabsf(xn[j].w - v.w);
    }
    float dist = (d0 + d1) + (d2 + d3);
    // exp(-dist) = exp2(-dist * log2(e)) -> v_mul + v_exp_f32
    acc += __builtin_amdgcn_exp2f(dist * -1.44269504088896340736f);
  }

  out[(size_t)n * F_ + f] = acc;
}

extern "C" void kernel_launch(void* const* d_in, const int* in_sizes, int n_in,
                              void* d_out, int out_size, void* d_ws, size_t ws_size,
                              hipStream_t stream) {
  (void)in_sizes; (void)n_in; (void)out_size; (void)d_ws; (void)ws_size;
  const float* x = (const float*)d_in[0];
  float* out     = (float*)d_out;
  dim3 grid(F_, N_ / NTILE, 1); // 128 x 2 blocks
  dim3 block(NTILE, 1, 1);      // 128 threads = 4 waves (wave32)
  MinibatchDisc_kernel<<<grid, block, 0, stream>>>(x, out);
}